// TemporalGNN_47940424958298
// MI455X (gfx1250) — compile-verified
//
#include <hip/hip_runtime.h>
#include <math.h>

// Problem constants (from reference)
#define NN   5000      // nodes
#define FF   4         // features
#define TT   12        // periods
#define CC   600       // channels
#define EE   160000    // edges
#define FT   48        // F*T
#define CP   608       // C padded to 16
#define NP   5008      // N padded to 16
#define NTIL 313       // ceil(N/16)
#define DTIL 38        // ceil(C/16)

typedef __attribute__((ext_vector_type(2))) float v2f;
typedef __attribute__((ext_vector_type(8))) float v8f;
typedef __attribute__((ext_vector_type(4))) int   v4i;

#if defined(__gfx1250__) && __has_builtin(__builtin_amdgcn_global_load_async_to_lds_b128) && __has_builtin(__builtin_amdgcn_s_wait_asynccnt)
#define USE_ASYNC_LDS 1
typedef __attribute__((address_space(1))) v4i gv4i;   // global int4
typedef __attribute__((address_space(3))) v4i lv4i;   // LDS int4
#else
#define USE_ASYNC_LDS 0
#endif

// ---------------- workspace layout (float offsets) ----------------
#define WS_PROBS 0         // 16
#define WS_MZ    64        // CP*4 = 2432
#define WS_MH    2496      // 2432
#define WS_CZ    4928      // 608
#define WS_CH    5536      // 608
#define WS_DEG   6144      // 5008
#define WS_DIS   11152     // 5008
#define WS_AGG   16160     // NP*FT = 240384
#define WS_H     256544    // NP*CP = 3044864  (total ~13.2 MB)

// ---------------- 1) fold gate linears into the F->C conv --------------
// Mz[d,f] = sum_c Wlz[d, c] * W_z[f, c]   (only first C cols of Wlz alive)
// cz[d]   = blz[d] + sum_c Wlz[d,c]*b_z[c];   same for h-gate.  Plus softmax(att).
__global__ void k_precompute(const float* __restrict__ Wz, const float* __restrict__ bz,
                             const float* __restrict__ Wh, const float* __restrict__ bh,
                             const float* __restrict__ Wlz, const float* __restrict__ blz,
                             const float* __restrict__ Wlh, const float* __restrict__ blh,
                             const float* __restrict__ att,
                             float* __restrict__ Mz, float* __restrict__ Mh,
                             float* __restrict__ cz, float* __restrict__ ch,
                             float* __restrict__ probs)
{
    int gid = blockIdx.x * blockDim.x + threadIdx.x;
    const int nM = CP * FF;
    if (gid < nM) {                                  // Mz entry
        int d = gid / FF, f = gid % FF;
        float s = 0.f;
        if (d < CC) for (int c = 0; c < CC; ++c) s += Wlz[d * 2 * CC + c] * Wz[f * CC + c];
        Mz[d * FF + f] = s;
    } else if (gid < 2 * nM) {                       // Mh entry
        int g = gid - nM; int d = g / FF, f = g % FF;
        float s = 0.f;
        if (d < CC) for (int c = 0; c < CC; ++c) s += Wlh[d * 2 * CC + c] * Wh[f * CC + c];
        Mh[d * FF + f] = s;
    } else if (gid < 2 * nM + CP) {                  // cz
        int d = gid - 2 * nM;
        float s = 0.f;
        if (d < CC) { s = blz[d]; for (int c = 0; c < CC; ++c) s += Wlz[d * 2 * CC + c] * bz[c]; }
        cz[d] = s;
    } else if (gid < 2 * nM + 2 * CP) {              // ch
        int d = gid - 2 * nM - CP;
        float s = 0.f;
        if (d < CC) { s = blh[d]; for (int c = 0; c < CC; ++c) s += Wlh[d * 2 * CC + c] * bh[c]; }
        ch[d] = s;
    } else if (gid == 2 * nM + 2 * CP) {             // softmax(att), T=12
        float m = -1e30f;
        for (int t = 0; t < TT; ++t) m = fmaxf(m, att[t]);
        float e[TT], s = 0.f;
        for (int t = 0; t < TT; ++t) { e[t] = __expf(att[t] - m); s += e[t]; }
        float inv = 1.0f / s;
        for (int t = 0; t < TT; ++t) probs[t] = e[t] * inv;
    }
}

// ---------------- 2) GCN normalization ----------------
__global__ void k_init_deg(float* __restrict__ deg)
{
    int n = blockIdx.x * blockDim.x + threadIdx.x;
    if (n < NP) deg[n] = 1.0f;              // self-loop weight
}

__global__ void k_accum_deg(const int* __restrict__ ei, const float* __restrict__ ew,
                            float* __restrict__ deg)
{
    int e = blockIdx.x * blockDim.x + threadIdx.x;
    if (e < EE) atomicAdd(&deg[ei[EE + e]], ew[e]);   // col = edge_index[1]
}

__global__ void k_dis(const float* __restrict__ deg, float* __restrict__ dis)
{
    int n = blockIdx.x * blockDim.x + threadIdx.x;
    if (n < NP) dis[n] = deg[n] > 0.f ? rsqrtf(deg[n]) : 0.f;
}

// agg := self-loop contribution dis[n]^2 * x[n]; zero the padded node rows
__global__ void k_selfloop(const float* __restrict__ x, const float* __restrict__ dis,
                           float* __restrict__ agg)
{
    int gid = blockIdx.x * blockDim.x + threadIdx.x;
    if (gid >= NP * FT) return;
    int n = gid / FT, k = gid % FT;
    agg[gid] = (n < NN) ? dis[n] * dis[n] * x[n * FT + k] : 0.f;
}

// edge scatter: agg[col] += dis[row]*w*dis[col] * x[row]   (48 floats / edge)
__global__ void k_scatter(const float* __restrict__ x, const int* __restrict__ ei,
                          const float* __restrict__ ew, const float* __restrict__ dis,
                          float* __restrict__ agg)
{
    int gid = blockIdx.x * blockDim.x + threadIdx.x;
    if (gid >= EE * FT) return;
    int e = gid / FT, k = gid % FT;
    int r = ei[e], c = ei[EE + e];
    float nrm = dis[r] * ew[e] * dis[c];
    atomicAdd(&agg[c * FT + k], nrm * x[r * FT + k]);
}

// ---------------- 3) gates + temporal attention, WMMA f32 16x16x4 ----------
// One block = 16-node tile staged in LDS; 8 waves cover 8 d-tiles (grid.y=5).
// Per t: pre = A(16 nodes x 4 f) * B(4 f x 16 d) via V_WMMA_F32_16X16X4_F32,
// shared A, two B operands (z- and h-gate). Accumulate probs[t]*(1-sig)*tanh.
__global__ void __launch_bounds__(256) k_gate(const float* __restrict__ agg,
                                              const float* __restrict__ Mz,
                                              const float* __restrict__ Mh,
                                              const float* __restrict__ cz,
                                              const float* __restrict__ ch,
                                              const float* __restrict__ probs,
                                              float* __restrict__ H)
{
    __shared__ float sAgg[16 * 49];   // 16 nodes x 48 (padded to 49 vs 64 banks)
    __shared__ float sP[TT];
    const int tid = threadIdx.x;
    const int n0 = blockIdx.x * 16;

#if USE_ASYNC_LDS
    // Stage the 16x48 agg tile via GLOBAL_LOAD_ASYNC_TO_LDS_B128 (ASYNCcnt path).
    // 192 lanes move one float4 each; rows are 48 floats (multiple of 4) so a
    // b128 never crosses the padded LDS row stride of 49.
    if (tid < 192) {
        int m = tid / 12, q = (tid % 12) * 4;
        __builtin_amdgcn_global_load_async_to_lds_b128(
            (gv4i*)(agg + (size_t)(n0 + m) * FT + q),
            (lv4i*)&sAgg[m * 49 + q],
            0, 0);
    }
    if (tid < TT) sP[tid] = probs[tid];
    __builtin_amdgcn_s_wait_asynccnt(0);
    __syncthreads();
#else
    for (int i = tid; i < 16 * FT; i += 256) {
        int m = i / FT, k = i % FT;
        sAgg[m * 49 + k] = agg[n0 * FT + i];
    }
    if (tid < TT) sP[tid] = probs[tid];
    __syncthreads();
#endif

    const int wave = tid >> 5, lane = tid & 31;
    const int dtile = blockIdx.y * 8 + wave;
    if (dtile >= DTIL) return;                 // uniform per-wave: EXEC stays full

    const int ln = lane & 15, hi = lane >> 4, kb = hi * 2;
    const int d = dtile * 16 + ln;

    // B operands (t-invariant): B[K=f][N=d] = M[d,f]; lane holds K={kb,kb+1}
    v2f Bz, Bh;
    Bz.x = Mz[d * FF + kb];  Bz.y = Mz[d * FF + kb + 1];
    Bh.x = Mh[d * FF + kb];  Bh.y = Mh[d * FF + kb + 1];
    const float czd = cz[d], chd = ch[d];

    float hacc[8];
#pragma unroll
    for (int v = 0; v < 8; ++v) hacc[v] = 0.f;

    // Unroll capped at 4: enough WMMAs in flight to overlap with the previous
    // iteration's exp/rcp chains, without spilling past v255 (full unroll made
    // the allocator reach for s_set_vgpr_msb 1024-VGPR addressing and killed
    // occupancy with ~24 live 8-VGPR accumulators).
#pragma unroll 4
    for (int t = 0; t < TT; ++t) {
        // A[M=node][K=f]: lane(ln) rows, K columns {kb, kb+1}
        v2f A;
        A.x = sAgg[ln * 49 + kb * TT + t];
        A.y = sAgg[ln * 49 + (kb + 1) * TT + t];
        v8f c0 = {};
        v8f zp = __builtin_amdgcn_wmma_f32_16x16x4_f32(false, A, false, Bz,
                                                       (short)0, c0, false, false);
        v8f hp = __builtin_amdgcn_wmma_f32_16x16x4_f32(false, A, false, Bh,
                                                       (short)0, c0, false, false);
        const float pt = sP[t];
#pragma unroll
        for (int v = 0; v < 8; ++v) {
            // Branch-free activations (no EXEC-divergent libm paths near WMMA):
            //   1 - sigmoid(x) = 1/(1+exp(x));   tanh(x) = 2/(1+exp(-2x)) - 1.
            // v_exp_f32 saturates cleanly at both extremes, so limits are exact.
            float omz = 1.0f / (1.0f + __expf(zp[v] + czd));
            float ht  = 2.0f / (1.0f + __expf(-2.0f * (hp[v] + chd))) - 1.0f;
            hacc[v] += pt * omz * ht;             // H_t = (1-z)*h_tilde, z*H == 0
        }
    }

    // D layout: lane owns column d, rows v + 8*hi.  Store relu(H_accum).
#pragma unroll
    for (int v = 0; v < 8; ++v) {
        int node = n0 + v + 8 * hi;
        H[(size_t)node * CP + d] = fmaxf(hacc[v], 0.f);
    }
}

// ---------------- 4) head: out[n,t'] = relu(H)[n,:] . W_head[t',:] + b -------
__global__ void k_head(const float* __restrict__ H, const float* __restrict__ Wh,
                       const float* __restrict__ bh, float* __restrict__ out)
{
    int gid = blockIdx.x * blockDim.x + threadIdx.x;
    if (gid >= NN * TT) return;
    int n = gid / TT, tp = gid % TT;
    const float* h = H + (size_t)n * CP;
    const float* w = Wh + tp * CC;
    float s = bh[tp];
    for (int dd = 0; dd < CC; ++dd) s += h[dd] * w[dd];
    out[n * TT + tp] = s;
}

extern "C" void kernel_launch(void* const* d_in, const int* in_sizes, int n_in,
                              void* d_out, int out_size, void* d_ws, size_t ws_size,
                              hipStream_t stream)
{
    (void)in_sizes; (void)n_in; (void)out_size; (void)ws_size;
    const float* x     = (const float*)d_in[0];
    const int*   ei    = (const int*)  d_in[1];
    const float* ew    = (const float*)d_in[2];
    const float* Wz    = (const float*)d_in[3];
    const float* bz    = (const float*)d_in[4];
    // d_in[5], d_in[6]: W_r, b_r — exactly dead (H == 0 every period)
    const float* Wh    = (const float*)d_in[7];
    const float* bhc   = (const float*)d_in[8];
    const float* Wlz   = (const float*)d_in[9];
    const float* blz   = (const float*)d_in[10];
    // d_in[11], d_in[12]: Wlr, blr — dead
    const float* Wlh   = (const float*)d_in[13];
    const float* blh   = (const float*)d_in[14];
    const float* att   = (const float*)d_in[15];
    const float* Whead = (const float*)d_in[16];
    const float* bhead = (const float*)d_in[17];

    float* ws    = (float*)d_ws;
    float* probs = ws + WS_PROBS;
    float* Mz    = ws + WS_MZ;
    float* Mh    = ws + WS_MH;
    float* cz    = ws + WS_CZ;
    float* ch    = ws + WS_CH;
    float* deg   = ws + WS_DEG;
    float* dis   = ws + WS_DIS;
    float* agg   = ws + WS_AGG;
    float* H     = ws + WS_H;

    // fold gate linears + softmax (tiny)
    {
        int nthr = 2 * CP * FF + 2 * CP + 1;
        k_precompute<<<(nthr + 255) / 256, 256, 0, stream>>>(
            Wz, bz, Wh, bhc, Wlz, blz, Wlh, blh, att, Mz, Mh, cz, ch, probs);
    }
    // GCN degree / normalization
    k_init_deg<<<(NP + 255) / 256, 256, 0, stream>>>(deg);
    k_accum_deg<<<(EE + 255) / 256, 256, 0, stream>>>(ei, ew, deg);
    k_dis<<<(NP + 255) / 256, 256, 0, stream>>>(deg, dis);
    // propagate: self loops then edge scatter (bandwidth-dominant step)
    k_selfloop<<<(NP * FT + 255) / 256, 256, 0, stream>>>(x, dis, agg);
    k_scatter<<<(EE * FT + 255) / 256, 256, 0, stream>>>(x, ei, ew, dis, agg);
    // WMMA gate + temporal attention
    {
        dim3 grid(NTIL, (DTIL + 7) / 8);   // 313 x 5
        k_gate<<<grid, 256, 0, stream>>>(agg, Mz, Mh, cz, ch, probs, H);
    }
    // head GEMV
    k_head<<<(NN * TT + 255) / 256, 256, 0, stream>>>(H, Whead, bhead, (float*)d_out);
}